// PytorchCircuit_13915694039379
// MI455X (gfx1250) — compile-verified
//
#include <hip/hip_runtime.h>
#include <hip/hip_bf16.h>
#include <stdint.h>

// Problem geometry: N = 2^22 amplitudes, 2 QAOA layers.
#define NQ     22
#define NTOT   (1u << NQ)
#define LOWB   12                 // bits 0..11 handled on contiguous chunks
#define LOWN   (1u << LOWB)       // 4096 elements / block (32 KB LDS re+im)
#define HIGHB  10                 // bits 12..21 handled on strided tiles
#define HIGHN  (1u << HIGHB)      // 1024 high positions
#define TPB    256
#define INV_SQRT_N (1.0f / 2048.0f)   // 1/sqrt(2^22)

typedef float v2f __attribute__((ext_vector_type(2)));
typedef float v8f __attribute__((ext_vector_type(8)));

#if defined(__gfx1250__) && __has_builtin(__builtin_amdgcn_wmma_f32_16x16x4_f32)
#define HAVE_WMMA_F32 1
#else
#define HAVE_WMMA_F32 0
#endif

// ---------------- async global<->LDS helpers (CDNA5 ASYNCcnt path) ----------
__device__ __forceinline__ void async_load_f4(float* lds_dst, const float* gsrc) {
#if defined(__gfx1250__)
    uint32_t l = (uint32_t)(uintptr_t)lds_dst;   // low 32 bits of generic ptr = LDS offset
    asm volatile("global_load_async_to_lds_b128 %0, %1, off"
                 :: "v"(l), "v"(gsrc) : "memory");
#else
    *(float4*)lds_dst = *(const float4*)gsrc;
#endif
}

__device__ __forceinline__ void async_store_f4(float* gdst, const float* lds_src) {
#if defined(__gfx1250__)
    uint32_t l = (uint32_t)(uintptr_t)lds_src;
    asm volatile("global_store_async_from_lds_b128 %0, %1, off"
                 :: "v"(gdst), "v"(l) : "memory");
#else
    *(float4*)gdst = *(const float4*)lds_src;
#endif
}

__device__ __forceinline__ void wait_async() {
#if defined(__gfx1250__)
    asm volatile("s_wait_asynccnt 0" ::: "memory");
#endif
}

// ---------------- butterfly: c' = cb*c + i*sb*flip(c) -----------------------
__device__ __forceinline__ void bf(float* lre, float* lim,
                                   unsigned lo, unsigned hi, float cb, float sb) {
    float are = lre[lo], aim = lim[lo];
    float bre = lre[hi], bim = lim[hi];
    lre[lo] = fmaf(cb, are, -sb * bim);
    lim[lo] = fmaf(cb, aim,  sb * bre);
    lre[hi] = fmaf(cb, bre, -sb * aim);
    lim[hi] = fmaf(cb, bim,  sb * are);
}

#if HAVE_WMMA_F32

#define WMMA_F32(A, B, C) \
    __builtin_amdgcn_wmma_f32_16x16x4_f32(false, (A), false, (B), (short)0, (C), false, false)

// Branchless 5-way select (lowers to v_cndmask, no EXEC divergence).
__device__ __forceinline__ float sel5(unsigned w, float a0, float a1, float a2,
                                      float a3, float a4) {
    float r = a0;
    r = (w == 1u) ? a1 : r;
    r = (w == 2u) ? a2 : r;
    r = (w == 3u) ? a3 : r;
    r = (w == 4u) ? a4 : r;
    return r;
}

// Precompute (once per kernel) this lane's B-matrix fragments for
// U4 = (cb*I + i*sb*X)^{tensor 4}: U4[r][c] = cb^(4-w) * (i*sb)^w, w = popc(r^c).
// B layout (4x16): v0 = K row {0|2} across lane halves, v1 = K row {1|3}.
__device__ __forceinline__ void make_b(float cb, float sb,
                                       v2f* bR, v2f* bI, v2f* bmI) {
    const unsigned lane   = threadIdx.x & 31;
    const bool     hihalf = lane >= 16;
    const unsigned n      = lane & 15;
    const float c2 = cb * cb, c3 = c2 * cb, c4 = c3 * cb;
    const float s2 = sb * sb, s3 = s2 * sb, s4 = s3 * sb;
    const float m0 = c4, m1 = c3 * sb, m2 = c2 * s2, m3 = cb * s3, m4 = s4;
    #pragma unroll
    for (unsigned kc = 0; kc < 4; ++kc) {
        const unsigned k0 = kc * 4 + (hihalf ? 2u : 0u);
        float re[2], im[2];
        #pragma unroll
        for (int j = 0; j < 2; ++j) {
            const unsigned w   = __popc(n ^ (k0 + (unsigned)j));
            const float    mag = sel5(w, m0, m1, m2, m3, m4);
            const float    sgn = 1.0f - (float)(w & 2u);   // i^w sign: +1 / -1
            const float    odd = (float)(w & 1u);          // 0 -> real, 1 -> imag
            const float    v   = mag * sgn;
            re[j] = v * (1.0f - odd);
            im[j] = v * odd;
        }
        bR[kc]  = (v2f){ re[0],  re[1] };
        bI[kc]  = (v2f){ im[0],  im[1] };
        bmI[kc] = (v2f){-im[0], -im[1] };
    }
}

// Apply 4 mixing stages (axis bits [P, P+4)) over a 4096-element LDS chunk as a
// batched complex 16x16 GEMM: out = S @ U4^T. 16 tiles of (16 rows x 16 axis),
// 8 waves x 2 tiles; per tile 16 chained V_WMMA_F32_16X16X4_F32.
// Element index: ((M>>P)<<(P+4)) | (k<<P) | (M & (2^P - 1)).
template <int P>
__device__ __forceinline__ void apply_u4_pass(float* lre, float* lim,
                                              const v2f* bR, const v2f* bI,
                                              const v2f* bmI) {
    const unsigned tid    = threadIdx.x;
    const unsigned lane   = tid & 31;
    const unsigned wave   = tid >> 5;
    const bool     hihalf = lane >= 16;
    const unsigned m      = lane & 15;          // A-matrix row / D-matrix col
    const unsigned LOWMASK = (1u << P) - 1u;

    #pragma unroll
    for (unsigned t = wave * 2; t < wave * 2 + 2; ++t) {
        const unsigned M      = t * 16 + m;
        const unsigned base_m = ((M >> P) << (P + 4)) | (M & LOWMASK);
        v8f accRe = {0, 0, 0, 0, 0, 0, 0, 0};
        v8f accIm = {0, 0, 0, 0, 0, 0, 0, 0};
        #pragma unroll
        for (unsigned kc = 0; kc < 4; ++kc) {
            const unsigned k0 = kc * 4 + (hihalf ? 2u : 0u);
            v2f ar, ai;
            ar.x = lre[base_m + (k0 << P)];
            ar.y = lre[base_m + ((k0 + 1) << P)];
            ai.x = lim[base_m + (k0 << P)];
            ai.y = lim[base_m + ((k0 + 1) << P)];
            // re' = Sre@R^T - Sim@I^T ; im' = Sre@I^T + Sim@R^T
            accRe = WMMA_F32(ar, bR[kc],  accRe);
            accRe = WMMA_F32(ai, bmI[kc], accRe);
            accIm = WMMA_F32(ar, bI[kc],  accIm);
            accIm = WMMA_F32(ai, bR[kc],  accIm);
        }
        // D layout: vgpr v -> (M = v + (lane>=16 ? 8 : 0), N = lane&15)
        #pragma unroll
        for (unsigned v = 0; v < 8; ++v) {
            const unsigned Mv  = t * 16 + v + (hihalf ? 8u : 0u);
            const unsigned bmv = ((Mv >> P) << (P + 4)) | (Mv & LOWMASK);
            lre[bmv + (m << P)] = accRe[v];
            lim[bmv + (m << P)] = accIm[v];
        }
    }
}
#endif // HAVE_WMMA_F32

// =============== Kernel 1: init |+>^n, layer-0 phase, layer-0 low stages ====
__global__ __launch_bounds__(TPB)
void k_init_low(const float* __restrict__ h,
                const float* __restrict__ gamma, const float* __restrict__ beta,
                float* __restrict__ wre, float* __restrict__ wim) {
    __shared__ __align__(16) float lre[LOWN];
    __shared__ __align__(16) float lim[LOWN];
    const unsigned tid  = threadIdx.x;
    const size_t   base = (size_t)blockIdx.x * LOWN;
    const float g0 = gamma[0];
    const float b0 = beta[0];
    const float cb = __cosf(b0), sb = __sinf(b0);

    #pragma unroll
    for (unsigned k = 0; k < LOWN / TPB; ++k) {
        const unsigned idx = tid + k * TPB;
        const float ang = g0 * h[base + idx];
        lre[idx] = __cosf(ang) * INV_SQRT_N;
        lim[idx] = __sinf(ang) * INV_SQRT_N;
    }
    __syncthreads();

#if HAVE_WMMA_F32
    v2f bR[4], bI[4], bmI[4];
    make_b(cb, sb, bR, bI, bmI);
    apply_u4_pass<0>(lre, lim, bR, bI, bmI);   // bits 0..3
    __syncthreads();
    apply_u4_pass<4>(lre, lim, bR, bI, bmI);   // bits 4..7
    __syncthreads();
    apply_u4_pass<8>(lre, lim, bR, bI, bmI);   // bits 8..11
    __syncthreads();
#else
    for (int b = 0; b < LOWB; ++b) {
        const unsigned stride = 1u << b;
        #pragma unroll
        for (unsigned p = 0; p < LOWN / 2 / TPB; ++p) {
            const unsigned q  = tid + p * TPB;
            const unsigned lo = ((q >> b) << (b + 1)) | (q & (stride - 1));
            bf(lre, lim, lo, lo + stride, cb, sb);
        }
        __syncthreads();
    }
#endif

    #pragma unroll
    for (unsigned k = 0; k < LOWN / TPB / 4; ++k) {
        const unsigned i4 = tid + k * TPB;
        *(float4*)&wre[base + (size_t)i4 * 4] = *(float4*)&lre[i4 * 4];
        *(float4*)&wim[base + (size_t)i4 * 4] = *(float4*)&lim[i4 * 4];
    }
}

// ===== Kernel 2: layer-0 high stages, layer-1 phase, layer-1 high stages ====
// Tile: all 1024 high positions x 4 contiguous low offsets (stride LOWN gather).
// In LDS, element index = hgh*4 + jj, so hgh bit b == element bit (b+2).
__global__ __launch_bounds__(TPB)
void k_high(const float* __restrict__ h,
            const float* __restrict__ gamma, const float* __restrict__ beta,
            float* __restrict__ wre, float* __restrict__ wim) {
    __shared__ __align__(16) float lre[HIGHN * 4];
    __shared__ __align__(16) float lim[HIGHN * 4];
    const unsigned tid = threadIdx.x;
    const unsigned j0  = blockIdx.x * 4;
    const float cb0 = __cosf(beta[0]), sb0 = __sinf(beta[0]);
    const float cb1 = __cosf(beta[1]), sb1 = __sinf(beta[1]);
    const float g1  = gamma[1];

    // async DMA fill: one b128 per (array,row)
    #pragma unroll
    for (unsigned r = 0; r < 4; ++r) {
        const unsigned hg = tid + r * TPB;
        const size_t goff = (size_t)hg * LOWN + j0;
        async_load_f4(&lre[hg * 4], &wre[goff]);
        async_load_f4(&lim[hg * 4], &wim[goff]);
    }
    wait_async();
    __syncthreads();

#if HAVE_WMMA_F32
    v2f b0R[4], b0I[4], b0mI[4];
    make_b(cb0, sb0, b0R, b0I, b0mI);
    apply_u4_pass<2>(lre, lim, b0R, b0I, b0mI);   // hgh bits 0..3
    __syncthreads();
    apply_u4_pass<6>(lre, lim, b0R, b0I, b0mI);   // hgh bits 4..7
    __syncthreads();
    const int hstart = 8;
#else
    const int hstart = 0;
#endif
    for (int b = hstart; b < HIGHB; ++b) {        // remaining hgh bits
        const unsigned stride = 1u << b;
        #pragma unroll
        for (unsigned p = 0; p < 8; ++p) {
            const unsigned q   = tid + p * TPB;   // 2048 pair-slots
            const unsigned jj  = q & 3, pq = q >> 2;
            const unsigned loh = ((pq >> b) << (b + 1)) | (pq & (stride - 1));
            bf(lre, lim, loh * 4 + jj, (loh + stride) * 4 + jj, cb0, sb0);
        }
        __syncthreads();
    }

    // layer-1 phase rotation (elementwise, after ALL layer-0 stages are done)
    #pragma unroll
    for (unsigned r = 0; r < 4; ++r) {
        const unsigned hg = tid + r * TPB;
        const float4 h4 = *(const float4*)&h[(size_t)hg * LOWN + j0];
        #pragma unroll
        for (int jj = 0; jj < 4; ++jj) {
            const float ang = g1 * ((const float*)&h4)[jj];
            const float ca = __cosf(ang), sa = __sinf(ang);
            const unsigned li = hg * 4 + jj;
            const float re = lre[li], im = lim[li];
            lre[li] = re * ca - im * sa;
            lim[li] = re * sa + im * ca;
        }
    }
    __syncthreads();

    // layer-1 stages on high bits (stages within a layer commute)
#if HAVE_WMMA_F32
    v2f b1R[4], b1I[4], b1mI[4];
    make_b(cb1, sb1, b1R, b1I, b1mI);
    apply_u4_pass<2>(lre, lim, b1R, b1I, b1mI);
    __syncthreads();
    apply_u4_pass<6>(lre, lim, b1R, b1I, b1mI);
    __syncthreads();
#endif
    for (int b = hstart; b < HIGHB; ++b) {
        const unsigned stride = 1u << b;
        #pragma unroll
        for (unsigned p = 0; p < 8; ++p) {
            const unsigned q   = tid + p * TPB;
            const unsigned jj  = q & 3, pq = q >> 2;
            const unsigned loh = ((pq >> b) << (b + 1)) | (pq & (stride - 1));
            bf(lre, lim, loh * 4 + jj, (loh + stride) * 4 + jj, cb1, sb1);
        }
        __syncthreads();
    }

    // async DMA drain
    #pragma unroll
    for (unsigned r = 0; r < 4; ++r) {
        const unsigned hg = tid + r * TPB;
        const size_t goff = (size_t)hg * LOWN + j0;
        async_store_f4(&wre[goff], &lre[hg * 4]);
        async_store_f4(&wim[goff], &lim[hg * 4]);
    }
    wait_async();
}

// ========= Kernel 3: layer-1 low stages, |amp|^2 . hS reduction =============
__global__ __launch_bounds__(TPB)
void k_low_final(const float* __restrict__ hS, const float* __restrict__ beta,
                 float* __restrict__ wre, float* __restrict__ wim,
                 float* __restrict__ out) {
    __shared__ __align__(16) float lre[LOWN];
    __shared__ __align__(16) float lim[LOWN];
    __shared__ float red[TPB];
    const unsigned tid  = threadIdx.x;
    const size_t   base = (size_t)blockIdx.x * LOWN;
    const float b1 = beta[1];
    const float cb = __cosf(b1), sb = __sinf(b1);

    #pragma unroll
    for (unsigned k = 0; k < LOWN / TPB / 4; ++k) {
        const unsigned i4 = tid + k * TPB;
        async_load_f4(&lre[i4 * 4], &wre[base + (size_t)i4 * 4]);
        async_load_f4(&lim[i4 * 4], &wim[base + (size_t)i4 * 4]);
    }
    wait_async();
    __syncthreads();

#if HAVE_WMMA_F32
    v2f bR[4], bI[4], bmI[4];
    make_b(cb, sb, bR, bI, bmI);
    apply_u4_pass<0>(lre, lim, bR, bI, bmI);
    __syncthreads();
    apply_u4_pass<4>(lre, lim, bR, bI, bmI);
    __syncthreads();
    apply_u4_pass<8>(lre, lim, bR, bI, bmI);
    __syncthreads();
#else
    for (int b = 0; b < LOWB; ++b) {
        const unsigned stride = 1u << b;
        #pragma unroll
        for (unsigned p = 0; p < LOWN / 2 / TPB; ++p) {
            const unsigned q  = tid + p * TPB;
            const unsigned lo = ((q >> b) << (b + 1)) | (q & (stride - 1));
            bf(lre, lim, lo, lo + stride, cb, sb);
        }
        __syncthreads();
    }
#endif

    float acc = 0.0f;
    #pragma unroll
    for (unsigned k = 0; k < LOWN / TPB; ++k) {
        const unsigned idx = tid + k * TPB;
        const float re = lre[idx], im = lim[idx];
        const float p = fmaf(re, re, im * im);
        acc = fmaf(p, hS[base + idx], acc);
    }
    red[tid] = acc;
    __syncthreads();
    for (unsigned s = TPB / 2; s > 0; s >>= 1) {
        if (tid < s) red[tid] += red[tid + s];
        __syncthreads();
    }
    if (tid == 0) atomicAdd(out, red[0]);
}

__global__ void k_zero(float* out, int n) {
    const int i = blockIdx.x * blockDim.x + threadIdx.x;
    if (i < n) out[i] = 0.0f;
}

// ============================= launcher =====================================
extern "C" void kernel_launch(void* const* d_in, const int* in_sizes, int n_in,
                              void* d_out, int out_size, void* d_ws, size_t ws_size,
                              hipStream_t stream) {
    const float* h     = (const float*)d_in[0];
    const float* hS    = (const float*)d_in[1];
    const float* gamma = (const float*)d_in[2];
    const float* beta  = (const float*)d_in[3];
    float* wre = (float*)d_ws;           // N floats (16 MB)
    float* wim = wre + NTOT;             // N floats (16 MB)
    float* out = (float*)d_out;

    k_zero<<<1, 32, 0, stream>>>(out, out_size);
    k_init_low <<<NTOT / LOWN, TPB, 0, stream>>>(h, gamma, beta, wre, wim);
    k_high     <<<LOWN / 4,    TPB, 0, stream>>>(h, gamma, beta, wre, wim);
    k_low_final<<<NTOT / LOWN, TPB, 0, stream>>>(hS, beta, wre, wim, out);
}